// TopKOp_27608049779406
// MI455X (gfx1250) — compile-verified
//
#include <hip/hip_runtime.h>

#define ROW_LEN 2048
#define WAVES_PER_BLOCK 8
#define THREADS (WAVES_PER_BLOCK * 32)
#define TOPK 10

typedef int v4i __attribute__((ext_vector_type(4)));
typedef v4i __attribute__((address_space(1))) v4i_global;
typedef v4i __attribute__((address_space(3))) v4i_lds;

// One wave32 per row of 2048 f32. Stage row to LDS (async on CDNA5), build
// per-lane sorted top-10 with a v_med3_f32 insertion chain, wave-merge via
// 10 rounds of max+pop, then mask.
__global__ __launch_bounds__(THREADS) void topk_mask_kernel(
    const float* __restrict__ x, float* __restrict__ out, int nrows) {
  __shared__ float sh[WAVES_PER_BLOCK * ROW_LEN];

  const int wave = threadIdx.x >> 5;
  const int lane = threadIdx.x & 31;
  const int row  = blockIdx.x * WAVES_PER_BLOCK + wave;
  if (row >= nrows) return;

  const float* __restrict__ gsrc = x   + (size_t)row * ROW_LEN;
  float* __restrict__       gdst = out + (size_t)row * ROW_LEN;
  float* shrow = sh + wave * ROW_LEN;

  const float NEG_INF = -__builtin_inff();

  // ---- Phase 1: stage row into LDS ------------------------------------
#if defined(__gfx1250__) && __has_builtin(__builtin_amdgcn_global_load_async_to_lds_b128)
  {
#pragma unroll
    for (int i = 0; i < 16; ++i) {
      const int elt = lane * 4 + i * 128;  // float index; 16B per lane per issue
      __builtin_amdgcn_global_load_async_to_lds_b128(
          (v4i_global*)(gsrc + elt),
          (v4i_lds*)(shrow + elt),
          /*offset=*/0, /*cpol=*/0);
    }
#if __has_builtin(__builtin_amdgcn_s_wait_asynccnt)
    __builtin_amdgcn_s_wait_asynccnt(0);
#else
    asm volatile("s_wait_asynccnt 0" ::: "memory");
#endif
  }
#else
  {
#pragma unroll
    for (int i = 0; i < 16; ++i) {
      const int elt = lane * 4 + i * 128;
      *(float4*)(shrow + elt) = *(const float4*)(gsrc + elt);
    }
  }
#endif
  // No barrier: each wave reads only the LDS region it wrote.

  // ---- Phase 2: per-lane sorted (descending) top-10 -------------------
  float t[TOPK];
#pragma unroll
  for (int j = 0; j < TOPK; ++j) t[j] = NEG_INF;

#pragma unroll
  for (int i = 0; i < 16; ++i) {
    const int elt = lane * 4 + i * 128;
    float4 v4 = *(const float4*)(shrow + elt);
    float vv[4] = {v4.x, v4.y, v4.z, v4.w};
#pragma unroll
    for (int e = 0; e < 4; ++e) {
      float v = vv[e];
      // Insert v into sorted-descending t[]; smallest falls off the end.
      // u[j] = max(t[j], min(t[j-1], v)) == med3(t[j-1], t[j], v) since
      // t[j-1] >= t[j]. Update j=9..1 (reads still-old t[j-1]), then head.
#if defined(__gfx1250__) && __has_builtin(__builtin_amdgcn_fmed3f)
#pragma unroll
      for (int j = TOPK - 1; j >= 1; --j)
        t[j] = __builtin_amdgcn_fmed3f(t[j - 1], t[j], v);
      t[0] = fmaxf(t[0], v);
#else
#pragma unroll
      for (int j = 0; j < TOPK; ++j) {
        float hi = fmaxf(t[j], v);
        v = fminf(t[j], v);
        t[j] = hi;
      }
#endif
    }
  }

  // ---- Phase 3: wave-wide kth via 10 rounds of max + single pop -------
  float kth = NEG_INF;
#pragma unroll 1
  for (int it = 0; it < TOPK; ++it) {
    float m = t[0];
#pragma unroll
    for (int off = 16; off >= 1; off >>= 1)
      m = fmaxf(m, __shfl_xor(m, off, 32));
    // Exactly one lane (lowest index holding m) pops its head; this keeps
    // duplicate-value multiplicity correct.
    unsigned long long b = __ballot(t[0] == m);
    int winner = __ffsll(b) - 1;
    if (lane == winner) {
#pragma unroll
      for (int j = 0; j < TOPK - 1; ++j) t[j] = t[j + 1];
      t[TOPK - 1] = NEG_INF;
    }
    kth = m;  // after the 10th round this is the 10th-largest value
  }

  // ---- Phase 4: mask and store ----------------------------------------
#pragma unroll
  for (int i = 0; i < 16; ++i) {
    const int elt = lane * 4 + i * 128;
    float4 v = *(const float4*)(shrow + elt);
    v.x = (v.x >= kth) ? v.x : NEG_INF;
    v.y = (v.y >= kth) ? v.y : NEG_INF;
    v.z = (v.z >= kth) ? v.z : NEG_INF;
    v.w = (v.w >= kth) ? v.w : NEG_INF;
    *(float4*)(gdst + elt) = v;
  }
}

extern "C" void kernel_launch(void* const* d_in, const int* in_sizes, int n_in,
                              void* d_out, int out_size, void* d_ws, size_t ws_size,
                              hipStream_t stream) {
  (void)n_in; (void)d_ws; (void)ws_size; (void)out_size;
  const float* x = (const float*)d_in[0];
  float* out = (float*)d_out;
  const int nrows = in_sizes[0] / ROW_LEN;  // 2*16*2048 = 65536 rows
  const int blocks = (nrows + WAVES_PER_BLOCK - 1) / WAVES_PER_BLOCK;
  topk_mask_kernel<<<blocks, THREADS, 0, stream>>>(x, out, nrows);
}